// BidirectionalLoss_4836133175888
// MI455X (gfx1250) — compile-verified
//
#include <hip/hip_runtime.h>

#define B_ROWS 4096
#define C_COLS 8192
#define TPB    256
#define EPT    32   // elements per thread (C_COLS / TPB)
#define F4PT   8    // float4 loads per thread

typedef __attribute__((ext_vector_type(2))) float v2f;
typedef __attribute__((ext_vector_type(8))) float v8f;

__device__ __forceinline__ unsigned long long u64max(unsigned long long a, unsigned long long b) {
    return a > b ? a : b;
}

__device__ __forceinline__ unsigned long long shflxor64(unsigned long long v, int m) {
    unsigned lo = __shfl_xor((unsigned)(v & 0xFFFFFFFFull), m, 32);
    unsigned hi = __shfl_xor((unsigned)(v >> 32), m, 32);
    return ((unsigned long long)hi << 32) | lo;
}

// One workgroup per (row, direction). Fused BCE row-sum + exact top-6 hard-negative mining.
__global__ __launch_bounds__(TPB) void bidir_row_kernel(
    const float* __restrict__ tk_s, const float* __restrict__ g_s,
    const float* __restrict__ tk_t, const float* __restrict__ g_t,
    const float* __restrict__ conf, float* __restrict__ ws)
{
    const int row = blockIdx.x;
    const int d   = blockIdx.y;                 // 0 = tk, 1 = g
    const float* __restrict__ S = d ? g_s : tk_s;
    const float* __restrict__ T = d ? g_t : tk_t;
    const long long base = (long long)row * C_COLS;

    const int tid  = threadIdx.x;
    const int lane = tid & 31;
    const int wid  = tid >> 5;

    __shared__ unsigned long long sm_w[8];
    __shared__ unsigned long long sm_win[6];
    __shared__ float sm_f[8];

    // ---- streaming pass: BCE sum + cache scores in registers ----
    float vals[EPT];
    float lsum = 0.f;
    #pragma unroll
    for (int i = 0; i < F4PT; ++i) {
        const int col = i * (TPB * 4) + tid * 4;
        float4 p4 = *(const float4*)(S + base + col);
        float4 t4 = *(const float4*)(T + base + col);
        float pp[4] = {p4.x, p4.y, p4.z, p4.w};
        float tt[4] = {t4.x, t4.y, t4.z, t4.w};
        #pragma unroll
        for (int c = 0; c < 4; ++c) {
            const float p = pp[c];
            const float q = (tt[c] > 0.5f) ? p : (1.0f - p);  // t in {0,1} exactly
            lsum += -__logf(q);                               // one trans op / element
            vals[i * 4 + c] = p;
        }
    }

    // ---- row-sum tree reduction ----
    float s = lsum;
    #pragma unroll
    for (int off = 16; off; off >>= 1) s += __shfl_xor(s, off, 32);
    if (lane == 0) sm_f[wid] = s;

    // ---- per-thread argmax key: (value_bits << 32) | ~col  (lower col wins ties) ----
    unsigned long long myKey = 0ull; int myJ = 0;
    #pragma unroll
    for (int j = 0; j < EPT; ++j) {
        const int col = (j >> 2) * (TPB * 4) + tid * 4 + (j & 3);
        const unsigned long long k =
            ((unsigned long long)__float_as_uint(vals[j]) << 32) |
            (unsigned)(0xFFFFFFFFu - (unsigned)col);
        if (k > myKey) { myKey = k; myJ = j; }
    }
    __syncthreads();

    // ---- 6 rounds of exact workgroup argmax (iterative selection) ----
    for (int sel = 0; sel < 6; ++sel) {
        unsigned long long wmax = myKey;
        #pragma unroll
        for (int off = 16; off; off >>= 1) wmax = u64max(wmax, shflxor64(wmax, off));
        if (lane == 0) sm_w[wid] = wmax;
        __syncthreads();
        unsigned long long win = sm_w[0];
        #pragma unroll
        for (int k = 1; k < 8; ++k) win = u64max(win, sm_w[k]);
        if (tid == 0) sm_win[sel] = win;

        if (myKey == win) {  // unique winner (keys contain unique columns)
            #pragma unroll
            for (int j = 0; j < EPT; ++j) if (j == myJ) vals[j] = 0.0f; // < eps sentinel
            myKey = 0ull; myJ = 0;
            #pragma unroll
            for (int j = 0; j < EPT; ++j) {
                const int col = (j >> 2) * (TPB * 4) + tid * 4 + (j & 3);
                const unsigned long long k =
                    ((unsigned long long)__float_as_uint(vals[j]) << 32) |
                    (unsigned)(0xFFFFFFFFu - (unsigned)col);
                if (k > myKey) { myKey = k; myJ = j; }
            }
        }
        __syncthreads();
    }

    // ---- hard-negative selection among top-6 in rank order ----
    if (tid == 0) {
        float rowsum = 0.f;
        #pragma unroll
        for (int k = 0; k < 8; ++k) rowsum += sm_f[k];

        float negloss = 0.f; int cnt = 0;
        #pragma unroll
        for (int sel = 0; sel < 6; ++sel) {
            const unsigned long long k = sm_win[sel];
            const unsigned col = 0xFFFFFFFFu - (unsigned)(k & 0xFFFFFFFFull);
            const float p = __uint_as_float((unsigned)(k >> 32));
            const float t = T[base + col];
            if (t < 0.5f && cnt < 2) { negloss += -__logf(1.0f - p); ++cnt; }
        }
        // ws layout: [pos_tk | pos_g | negL_tk | negL_g | negC_tk | negC_g], each 4096 floats
        ws[(0 * 2 + d) * B_ROWS + row] = conf[row] * rowsum;
        ws[(1 * 2 + d) * B_ROWS + row] = negloss;
        ws[(2 * 2 + d) * B_ROWS + row] = (float)cnt;
    }
}

// Final cross-row reduction on the matrix pipe: ones(16x4) x B-chunk, C accumulates.
// Any bijective load of 64 elements into B gives sum(lanes, D[.][n] elem0) == 2 * chunk_sum.
__global__ __launch_bounds__(256) void bidir_finalize_kernel(
    const float* __restrict__ ws, float* __restrict__ out)
{
    const int tid  = threadIdx.x;
    const int lane = tid & 31;
    const int wid  = tid >> 5;
    __shared__ float sm_s[6];

    if (wid < 6) {  // wave-uniform branch -> EXEC all ones inside (WMMA requirement)
        const float* __restrict__ arr = ws + wid * B_ROWS;
        v2f a; a.x = 1.0f; a.y = 1.0f;
        v8f c = {};
        for (int base = 0; base < B_ROWS; base += 64) {
            v2f b;
            b.x = arr[base + lane];
            b.y = arr[base + 32 + lane];
            c = __builtin_amdgcn_wmma_f32_16x16x4_f32(
                    false, a, false, b, (short)0, c, false, false);
        }
        float tot = c[0];
        #pragma unroll
        for (int off = 16; off; off >>= 1) tot += __shfl_xor(tot, off, 32);
        tot *= 0.5f;  // rows 0 and 8 of D both hold the column sums
        if (lane == 0) sm_s[wid] = tot;
    }
    __syncthreads();

    if (tid == 0) {
        const float invBC = 1.0f / ((float)B_ROWS * (float)C_COLS);
        const float tk = sm_s[0] * invBC + 0.5f * sm_s[2] / (sm_s[4] + 1e-8f);
        const float g  = sm_s[1] * invBC + 0.5f * sm_s[3] / (sm_s[5] + 1e-8f);
        out[0] = 0.6f * tk + 0.4f * g;
        out[1] = tk;
        out[2] = g;
    }
}

extern "C" void kernel_launch(void* const* d_in, const int* in_sizes, int n_in,
                              void* d_out, int out_size, void* d_ws, size_t ws_size,
                              hipStream_t stream) {
    const float* tk_s = (const float*)d_in[0];
    const float* g_s  = (const float*)d_in[1];
    const float* tk_t = (const float*)d_in[2];
    const float* g_t  = (const float*)d_in[3];
    const float* conf = (const float*)d_in[4];
    float* ws  = (float*)d_ws;   // needs 6 * 4096 * 4 = 96 KB
    float* out = (float*)d_out;  // 3 floats: total, tk_loss, g_loss

    dim3 grid(B_ROWS, 2);
    bidir_row_kernel<<<grid, TPB, 0, stream>>>(tk_s, g_s, tk_t, g_t, conf, ws);
    bidir_finalize_kernel<<<1, 256, 0, stream>>>(ws, out);
}